// ContrastiveAddGNNLayer_40381282517158
// MI455X (gfx1250) — compile-verified
//
#include <hip/hip_runtime.h>
#include <hip/hip_bf16.h>
#include <math.h>

// ---------------- problem constants ----------------
#define NN      50000      // nodes
#define NE      800000     // edges
#define DF      128        // feature dim == UNITS
#define NHEADS  4
#define HU      32         // per-head units
#define NEG_SLOPE 0.2f
#define BN_EPS    0.001f

typedef __attribute__((ext_vector_type(2))) float v2f;
typedef __attribute__((ext_vector_type(8))) float v8f;

// ---------------- workspace layout (floats) ----------------
#define OFF_H    ((size_t)0)                         // h:   NN*128
#define OFF_UPD  (OFF_H   + (size_t)NN * DF)         // upd: NN*128
#define OFF_AGG  (OFF_UPD + (size_t)NN * DF)         // agg: NN*128
#define OFF_SI   (OFF_AGG + (size_t)NN * DF)         // s_i: NN*4
#define OFF_SJ   (OFF_SI  + (size_t)NN * NHEADS)     // s_j: NN*4
#define OFF_MAX  (OFF_SJ  + (size_t)NN * NHEADS)     // seg max: NN*4
#define OFF_DEN  (OFF_MAX + (size_t)NN * NHEADS)     // seg sum: NN*4
#define OFF_W    (OFF_DEN + (size_t)NN * NHEADS)     // edge logits/w: NE*4
#define OFF_MACC (OFF_W   + (size_t)NE * NHEADS)     // bn mean acc: 128
#define OFF_SQ   (OFF_MACC + (size_t)DF)             // bn sq acc:   128

// ---------------- helpers ----------------
__device__ __forceinline__ void atomicMaxFloat(float* addr, float val) {
    if (val >= 0.0f) {
        atomicMax((int*)addr, __float_as_int(val));
    } else {
        atomicMin((unsigned int*)addr, __float_as_uint(val));
    }
}

// ---------------- kernel 0: init workspace ----------------
__global__ void k_init(float* __restrict__ ws) {
    int i = blockIdx.x * blockDim.x + threadIdx.x;
    size_t nagg = (size_t)NN * DF;
    if ((size_t)i < nagg) ws[OFF_AGG + i] = 0.0f;
    if (i < NN * NHEADS) {
        ws[OFF_MAX + i] = -INFINITY;
        ws[OFF_DEN + i] = 0.0f;
    }
    if (i < DF) {
        ws[OFF_MACC + i] = 0.0f;
        ws[OFF_SQ + i]   = 0.0f;
    }
}

// ---------------- kernel 1: f32 WMMA GEMM (compile-time specialized) ----------------
// C[M x 128] = A[M x 128] * B[128 x 128] + bias  (optional relu)
// WLAYOUT==0: B row-major [k*128 + col]                       (Wm)
// WLAYOUT==1: B head-blocked Wk[head][k][u]:
//             [(col/32)*4096 + k*32 + (col%32)]               (Wk)
// Block = 256 threads (8 wave32) = one 16-row stripe; wave w -> cols [16w,16w+16).
template <int WLAYOUT, int DORELU>
__global__ __launch_bounds__(256) void k_gemm_wmma(
    const float* __restrict__ A, const float* __restrict__ B,
    const float* __restrict__ bias, float* __restrict__ C, int M)
{
    __shared__ float atile[16 * DF];   // 8 KB of the 320 KB/WGP LDS

    const int tid = threadIdx.x;
    const int m0  = blockIdx.x * 16;

    // stage A tile (coalesced, zero-pad rows >= M)
    #pragma unroll
    for (int i = 0; i < 8; ++i) {
        const int idx = tid + i * 256;
        const int r = idx >> 7, c = idx & (DF - 1);
        const int row = m0 + r;
        atile[idx] = (row < M) ? A[(size_t)row * DF + c] : 0.0f;
    }
    __syncthreads();

    const int lane = tid & 31;
    const int half = lane >> 4;        // 0: K+0/K+1, 1: K+2/K+3
    const int l    = lane & 15;
    const int n0   = (tid >> 5) * 16;  // wave id -> col tile
    const int col  = n0 + l;

    // per-lane B column base + K stride (both layouts linear in k)
    const int b_base   = (WLAYOUT == 0) ? col
                                        : ((col >> 5) * (DF * HU) + (col & 31));
    const int b_kstride = (WLAYOUT == 0) ? DF : HU;

    const float* __restrict__ arow = &atile[l * DF + 2 * half];
    const float* __restrict__ bcol = &B[b_base + (size_t)(2 * half) * b_kstride];

    v8f acc = {};
    #pragma unroll
    for (int kb = 0; kb < DF; kb += 4) {
        v2f a = *(const v2f*)(arow + kb);          // A[l][kb+2h], A[l][kb+2h+1]
        v2f b;
        b.x = bcol[(size_t)kb * b_kstride];        // B[kb+2h]  [col]
        b.y = bcol[(size_t)(kb + 1) * b_kstride];  // B[kb+2h+1][col]
        // D = A(16x4,f32) * B(4x16,f32) + C  -> v_wmma_f32_16x16x4_f32
        acc = __builtin_amdgcn_wmma_f32_16x16x4_f32(
                  false, a, false, b, (short)0, acc, false, false);
    }

    const float bv = bias[col];
    #pragma unroll
    for (int r = 0; r < 8; ++r) {
        float v = acc[r] + bv;
        if (DORELU) v = fmaxf(v, 0.0f);
        const int row = m0 + r + 8 * half;
        if (row < M) C[(size_t)row * DF + col] = v;
    }
}

// ---------------- kernel 2: per-node attention scores ----------------
// one block = one node, 128 threads; each head's 32 units == one wave32
__global__ __launch_bounds__(128) void k_scores(
    const float* __restrict__ h, const float* __restrict__ a_i,
    const float* __restrict__ a_j, float* __restrict__ s_i,
    float* __restrict__ s_j)
{
    const int node = blockIdx.x;
    const int t = threadIdx.x;                 // t = head*32 + u
    const float hv = h[(size_t)node * DF + t];
    float vi = hv * a_i[t];
    float vj = hv * a_j[t];
    #pragma unroll
    for (int off = 16; off > 0; off >>= 1) {
        vi += __shfl_xor(vi, off, 32);
        vj += __shfl_xor(vj, off, 32);
    }
    if ((t & 31) == 0) {
        const int hd = t >> 5;
        s_i[node * NHEADS + hd] = vi;
        s_j[node * NHEADS + hd] = vj;
    }
}

// ---------------- kernel 3: edge logits + segment max ----------------
__global__ void k_edge_logits_max(
    const int* __restrict__ ei, const float* __restrict__ s_i,
    const float* __restrict__ s_j, const float* __restrict__ a_e,
    const float* __restrict__ b_a, float* __restrict__ wbuf,
    float* __restrict__ maxb)
{
    const int idx = blockIdx.x * blockDim.x + threadIdx.x;
    if (idx >= NE * NHEADS) return;
    const int e  = idx >> 2;
    const int hd = idx & 3;
    const int recv = ei[2 * e];
    const int send = ei[2 * e + 1];
    float lg = s_i[recv * NHEADS + hd] + s_j[send * NHEADS + hd]
             + a_e[hd] + b_a[hd];
    lg = (lg > 0.0f) ? lg : NEG_SLOPE * lg;    // leaky relu
    wbuf[idx] = lg;
    atomicMaxFloat(&maxb[recv * NHEADS + hd], lg);
}

// ---------------- kernel 4: exp + segment sum ----------------
__global__ void k_edge_exp_sum(
    const int* __restrict__ ei, const float* __restrict__ maxb,
    float* __restrict__ wbuf, float* __restrict__ denom)
{
    const int idx = blockIdx.x * blockDim.x + threadIdx.x;
    if (idx >= NE * NHEADS) return;
    const int e  = idx >> 2;
    const int hd = idx & 3;
    const int recv = ei[2 * e];
    const float w = expf(wbuf[idx] - maxb[recv * NHEADS + hd]);
    wbuf[idx] = w;
    atomicAdd(&denom[recv * NHEADS + hd], w);
}

// ---------------- kernel 5: weighted message scatter-add ----------------
// one thread per (edge, feature); agg (25.6 MB) is L2-resident on MI455X
__global__ __launch_bounds__(256) void k_edge_scatter(
    const int* __restrict__ ei, const float* __restrict__ h,
    const float* __restrict__ wbuf, const float* __restrict__ denom,
    float* __restrict__ agg)
{
    const long long idx = (long long)blockIdx.x * 256 + threadIdx.x;
    if (idx >= (long long)NE * DF) return;
    const int e   = (int)(idx >> 7);
    const int col = (int)(idx & (DF - 1));
    const int hd  = col >> 5;
    const int recv = ei[2 * e];
    const int send = ei[2 * e + 1];
    const float alpha = wbuf[e * NHEADS + hd]
                      / (denom[recv * NHEADS + hd] + 1e-16f);
    atomicAdd(&agg[(size_t)recv * DF + col],
              h[(size_t)send * DF + col] * alpha);
}

// ---------------- kernel 6: batchnorm partial sums ----------------
#define BN_ROWS 128
__global__ __launch_bounds__(128) void k_bn_partial(
    const float* __restrict__ upd, float* __restrict__ macc,
    float* __restrict__ sqacc)
{
    const int col = threadIdx.x;
    const int r0  = blockIdx.x * BN_ROWS;
    float s = 0.0f, s2 = 0.0f;
    for (int r = 0; r < BN_ROWS; ++r) {
        const int row = r0 + r;
        if (row < NN) {
            const float v = upd[(size_t)row * DF + col];
            s += v; s2 += v * v;
        }
    }
    atomicAdd(&macc[col], s);
    atomicAdd(&sqacc[col], s2);
}

// ---------------- kernel 7: finalize ----------------
__global__ void k_finalize(
    const float* __restrict__ upd, const float* __restrict__ agg,
    const float* __restrict__ macc, const float* __restrict__ sqacc,
    const float* __restrict__ gamma, const float* __restrict__ beta,
    float* __restrict__ out)
{
    const long long idx = (long long)blockIdx.x * blockDim.x + threadIdx.x;
    if (idx >= (long long)NN * DF) return;
    const int col = (int)(idx & (DF - 1));
    const float inv_n = 1.0f / (float)NN;
    const float mean = macc[col] * inv_n;
    const float var  = sqacc[col] * inv_n - mean * mean;
    const float normed = gamma[col] * (upd[idx] - mean)
                       * rsqrtf(var + BN_EPS) + beta[col];
    out[idx] = normed + fmaxf(agg[idx], 0.0f);
}

// ---------------- launch ----------------
extern "C" void kernel_launch(void* const* d_in, const int* in_sizes, int n_in,
                              void* d_out, int out_size, void* d_ws, size_t ws_size,
                              hipStream_t stream) {
    const float* x     = (const float*)d_in[0];
    const int*   ei    = (const int*)  d_in[1];   // [NE][2] int32
    const float* Wk    = (const float*)d_in[2];   // [H][128][32]
    const float* bk    = (const float*)d_in[3];   // [H][32] == flat 128
    const float* a_i   = (const float*)d_in[4];   // [H][32]
    const float* a_j   = (const float*)d_in[5];
    const float* a_e   = (const float*)d_in[6];   // [H][1]
    const float* b_a   = (const float*)d_in[7];   // [H]
    const float* Wm    = (const float*)d_in[8];   // [128][128] row-major
    const float* bm    = (const float*)d_in[9];
    const float* gamma = (const float*)d_in[10];
    const float* beta  = (const float*)d_in[11];
    float* out = (float*)d_out;
    float* ws  = (float*)d_ws;

    float* h_buf  = ws + OFF_H;
    float* upd    = ws + OFF_UPD;
    float* agg    = ws + OFF_AGG;
    float* s_i    = ws + OFF_SI;
    float* s_j    = ws + OFF_SJ;
    float* maxb   = ws + OFF_MAX;
    float* denom  = ws + OFF_DEN;
    float* wbuf   = ws + OFF_W;
    float* macc   = ws + OFF_MACC;
    float* sqacc  = ws + OFF_SQ;

    // 0) init
    {
        const long long n = (long long)NN * DF;
        k_init<<<(int)((n + 255) / 256), 256, 0, stream>>>(ws);
    }
    // 1) h = x @ Wk + bk    (head-blocked B layout, no relu)
    const int gemm_blocks = (NN + 15) / 16;   // 3125
    k_gemm_wmma<1, 0><<<gemm_blocks, 256, 0, stream>>>(x, Wk, bk, h_buf, NN);
    // 2) upd = relu(x @ Wm + bm)   (row-major B)
    k_gemm_wmma<0, 1><<<gemm_blocks, 256, 0, stream>>>(x, Wm, bm, upd, NN);
    // 3) attention scores
    k_scores<<<NN, 128, 0, stream>>>(h_buf, a_i, a_j, s_i, s_j);
    // 4) edge logits + segment max
    {
        const int n = NE * NHEADS;
        k_edge_logits_max<<<(n + 255) / 256, 256, 0, stream>>>(
            ei, s_i, s_j, a_e, b_a, wbuf, maxb);
    }
    // 5) exp + segment sum
    {
        const int n = NE * NHEADS;
        k_edge_exp_sum<<<(n + 255) / 256, 256, 0, stream>>>(
            ei, maxb, wbuf, denom);
    }
    // 6) message scatter
    {
        const long long n = (long long)NE * DF;
        k_edge_scatter<<<(int)((n + 255) / 256), 256, 0, stream>>>(
            ei, h_buf, wbuf, denom, agg);
    }
    // 7) batchnorm partials
    k_bn_partial<<<(NN + BN_ROWS - 1) / BN_ROWS, 128, 0, stream>>>(
        upd, macc, sqacc);
    // 8) finalize
    {
        const long long n = (long long)NN * DF;
        k_finalize<<<(int)((n + 255) / 256), 256, 0, stream>>>(
            upd, agg, macc, sqacc, gamma, beta, out);
    }
}